// SelfAttention_35115652612510
// MI455X (gfx1250) — compile-verified
//
#include <hip/hip_runtime.h>
#include <math.h>

typedef __attribute__((ext_vector_type(2))) float v2f;
typedef __attribute__((ext_vector_type(8))) float v8f;

#define DDIM 512

// ---------------------------------------------------------------------------
// Kernel 1: fused rank-1-softmax attention.
// out_attn[bt,i] = sum_j (a_j / Z_j) * exp(a_j*b_i - m_j)
//   m_j = a_j>=0 ? a_j*max(b) : a_j*min(b)
//   Z_j = sum_i exp(a_j*b_i - m_j)
// One block (512 threads = 16 wave32) per batch row; a/b live in LDS.
// ---------------------------------------------------------------------------
__global__ __launch_bounds__(512) void attn_kernel(const float* __restrict__ a,
                                                   const float* __restrict__ b,
                                                   float* __restrict__ attn_out) {
    __shared__ float a_s[DDIM];
    __shared__ float b_s[DDIM];
    __shared__ float s_s[DDIM];   // a_j / Z_j
    __shared__ float m_s[DDIM];   // per-column softmax shift
    __shared__ float rmax[DDIM];
    __shared__ float rmin[DDIM];

    const int bt  = blockIdx.x;
    const int tid = threadIdx.x;

    const float av = a[(size_t)bt * DDIM + tid];
    const float bv = b[(size_t)bt * DDIM + tid];
    a_s[tid] = av;
    b_s[tid] = bv;
    rmax[tid] = bv;
    rmin[tid] = bv;
    __syncthreads();

    // tree reduction for max(b), min(b)
    for (int s = DDIM / 2; s > 0; s >>= 1) {
        if (tid < s) {
            rmax[tid] = fmaxf(rmax[tid], rmax[tid + s]);
            rmin[tid] = fminf(rmin[tid], rmin[tid + s]);
        }
        __syncthreads();
    }
    const float bmax = rmax[0];
    const float bmin = rmin[0];

    // pass 1: thread tid owns column j = tid, computes Z_j
    const float t = av;
    const float m = (t >= 0.0f) ? t * bmax : t * bmin;
    float z = 0.0f;
#pragma unroll 8
    for (int i = 0; i < DDIM; ++i) {
        z += __expf(fmaf(t, b_s[i], -m));   // v_exp_f32 on TRANS pipe
    }
    s_s[tid] = t / z;
    m_s[tid] = m;
    __syncthreads();

    // pass 2: thread tid owns row i = tid
    const float bi = bv;
    float acc = 0.0f;
#pragma unroll 8
    for (int j = 0; j < DDIM; ++j) {
        acc += s_s[j] * __expf(fmaf(a_s[j], bi, -m_s[j]));
    }
    attn_out[(size_t)bt * DDIM + tid] = acc;
}

// ---------------------------------------------------------------------------
// Kernel 2: out = a + attn @ W^T + bias via V_WMMA_F32_16X16X4_F32.
// One wave32 per 16x16 output tile; K-loop of 128 WMMAs (K=4 each).
// A 16x4 f32 layout: lanes 0-15 row M=0..15; VGPR0 holds K = 2*(lane/16),
// VGPR1 holds K = 2*(lane/16)+1.  B 4x16 mirrored (row per half-VGPR).
// C/D 16x16: VGPR r, lane l -> M = r + 8*(l/16), N = l%16.
// ---------------------------------------------------------------------------
__global__ __launch_bounds__(128) void fc_kernel(const float* __restrict__ attn,
                                                 const float* __restrict__ W,
                                                 const float* __restrict__ aIn,
                                                 const float* __restrict__ bias,
                                                 float* __restrict__ out,
                                                 int Brows) {
    const int wave = threadIdx.x >> 5;
    const int lane = threadIdx.x & 31;
    const int tile = blockIdx.x * 4 + wave;
    const int tiles_n = DDIM / 16;                // 32
    const int bm = tile / tiles_n;
    const int bn = tile % tiles_n;
    if (bm * 16 >= Brows) return;                 // wave-uniform: EXEC stays all-1s

    const int half = lane >> 4;                   // 0 or 1
    const int lrow = lane & 15;

    const float* Arow = attn + (size_t)(bm * 16 + lrow) * DDIM + 2 * half;
    const float* Brow = W    + (size_t)(bn * 16 + lrow) * DDIM + 2 * half;  // W[n, :]

    v8f c = {};
#pragma unroll 4
    for (int k = 0; k < DDIM; k += 4) {
        v2f avec = *(const v2f*)(Arow + k);
        v2f bvec = *(const v2f*)(Brow + k);
        // 8 args: (neg_a, A, neg_b, B, c_mod, C, reuse_a, reuse_b)
        c = __builtin_amdgcn_wmma_f32_16x16x4_f32(false, avec, false, bvec,
                                                  (short)0, c, false, false);
    }

    const int n = bn * 16 + lrow;
    const float bvl = bias[n];
#pragma unroll
    for (int r = 0; r < 8; ++r) {
        const int m = bm * 16 + r + 8 * half;
        out[(size_t)m * DDIM + n] = c[r] + aIn[(size_t)m * DDIM + n] + bvl;
    }
}

// ---------------------------------------------------------------------------
extern "C" void kernel_launch(void* const* d_in, const int* in_sizes, int n_in,
                              void* d_out, int out_size, void* d_ws, size_t ws_size,
                              hipStream_t stream) {
    const float* a    = (const float*)d_in[0];   // [B, D]
    const float* b    = (const float*)d_in[1];   // [B, D]
    const float* W    = (const float*)d_in[2];   // [D, D]
    const float* bias = (const float*)d_in[3];   // [D]
    float* out = (float*)d_out;

    const int D     = in_sizes[3];               // 512
    const int Brows = in_sizes[0] / D;           // 1024
    (void)D; (void)n_in; (void)out_size; (void)ws_size;

    float* attn = (float*)d_ws;                  // [Brows, DDIM] f32 scratch (2 MB)

    attn_kernel<<<Brows, 512, 0, stream>>>(a, b, attn);

    const int tiles = (Brows / 16) * (DDIM / 16);   // 2048 tiles, 1 wave each
    fc_kernel<<<tiles / 4, 128, 0, stream>>>(attn, W, a, bias, out, Brows);
}